// CenterLoss_6176162971973
// MI455X (gfx1250) — compile-verified
//
#include <hip/hip_runtime.h>

// CenterLoss for MI455X (gfx1250, wave32).
// dist_i = ||x_i - centers[labels_i]||^2  (identical math to the reference),
// clipped to [1e-12, 1e12], then averaged over the batch.
//
// Memory-bound (16 MB traffic, ~6 MFLOP): stream x and gathered center rows
// with b128 loads, VALU-accumulate squared diffs into the exact
// V_WMMA_F32_16X16X4_F32 A-matrix layout, and use one FP32 WMMA per wave
// (B = ones in column 0) to perform the 16-row cross-lane reduction.
// All reductions are fixed-order -> bitwise deterministic output.

typedef float v2f __attribute__((ext_vector_type(2)));
typedef float v8f __attribute__((ext_vector_type(8)));

#define FEAT_DIM 512
#define BATCH 4096
#define ROWS_PER_WAVE 16
#define WAVES_PER_BLOCK 4
#define ROWS_PER_BLOCK (ROWS_PER_WAVE * WAVES_PER_BLOCK)   // 64
#define NUM_BLOCKS (BATCH / ROWS_PER_BLOCK)                // 64

__global__ __launch_bounds__(WAVES_PER_BLOCK * 32)
void center_loss_partials(const float* __restrict__ x,
                          const int* __restrict__ labels,
                          const float* __restrict__ centers,
                          float* __restrict__ partials) {
    const int lane     = threadIdx.x & 31;
    const int wave     = threadIdx.x >> 5;
    const int rowLocal = lane & 15;   // M index within the 16-row tile
    const int g        = lane >> 4;   // K-pair group: 0 -> K{0,1}, 1 -> K{2,3}
    const int row      = blockIdx.x * ROWS_PER_BLOCK + wave * ROWS_PER_WAVE + rowLocal;

    const int label = labels[row];
    const float* __restrict__ xr = x + (size_t)row * FEAT_DIM;
    const float* __restrict__ cr = centers + (size_t)label * FEAT_DIM;

    // This lane accumulates two disjoint 128-column slices of its row:
    //   slice0 = [g*256, g*256+128)  -> A[r][2g+0]
    //   slice1 = [g*256+128, g*256+256) -> A[r][2g+1]
    // Across the two lane groups that covers all 512 columns exactly once.
    const int cbase = g * 256;
    const float4* __restrict__ x0 = (const float4*)(xr + cbase);
    const float4* __restrict__ c0 = (const float4*)(cr + cbase);
    const float4* __restrict__ x1 = (const float4*)(xr + cbase + 128);
    const float4* __restrict__ c1 = (const float4*)(cr + cbase + 128);

    __builtin_prefetch(cr + cbase, 0, 1);   // global_prefetch_b8 on the gathered row

    float s0 = 0.0f, s1 = 0.0f;
#pragma unroll 8
    for (int i = 0; i < 32; ++i) {          // 32 float4 = 128 columns per slice
        float4 a = x0[i], b = c0[i];
        float dx = a.x - b.x, dy = a.y - b.y, dz = a.z - b.z, dw = a.w - b.w;
        s0 = fmaf(dx, dx, fmaf(dy, dy, fmaf(dz, dz, fmaf(dw, dw, s0))));
        float4 a1 = x1[i], b1 = c1[i];
        float ex = a1.x - b1.x, ey = a1.y - b1.y, ez = a1.z - b1.z, ew = a1.w - b1.w;
        s1 = fmaf(ex, ex, fmaf(ey, ey, fmaf(ez, ez, fmaf(ew, ew, s1))));
    }

    // A-matrix (16x4 f32): lane<16 holds A[r][0..1], lane>=16 holds A[r][2..3].
    v2f A; A.x = s0; A.y = s1;
    // B-matrix (4x16 f32): column 0 all ones, everything else zero
    // -> D[r][0] = sum_k A[r][k] = full 512-col squared distance of row r.
    const float bval = (rowLocal == 0) ? 1.0f : 0.0f;
    v2f B; B.x = bval; B.y = bval;
    v8f C = {};
    C = __builtin_amdgcn_wmma_f32_16x16x4_f32(
        /*neg_a=*/false, A, /*neg_b=*/false, B,
        /*c_mod=*/(short)0, C, /*reuse_a=*/false, /*reuse_b=*/false);

    // D layout (16x16 f32 C/D): lane 0 (N=0) holds M=0..7 in C[0..7],
    // lane 16 holds M=8..15. Clip per-sample, then fixed-order 8-sum.
    float sum8 = 0.0f;
#pragma unroll
    for (int r = 0; r < 8; ++r) {
        float d = C[r];
        d = fminf(fmaxf(d, 1e-12f), 1e12f);
        sum8 += d;
    }

    __shared__ float lds[WAVES_PER_BLOCK * 2];
    if (rowLocal == 0) lds[wave * 2 + g] = sum8;   // lanes 0 and 16 of each wave
    __syncthreads();
    if (threadIdx.x == 0) {
        float t = 0.0f;
#pragma unroll
        for (int i = 0; i < WAVES_PER_BLOCK * 2; ++i) t += lds[i];
        partials[blockIdx.x] = t;
    }
}

__global__ void center_loss_final(const float* __restrict__ partials,
                                  float* __restrict__ out) {
    if (threadIdx.x == 0) {
        float t = 0.0f;
        for (int i = 0; i < NUM_BLOCKS; ++i) t += partials[i];   // fixed order
        out[0] = t * (1.0f / (float)BATCH);
    }
}

extern "C" void kernel_launch(void* const* d_in, const int* in_sizes, int n_in,
                              void* d_out, int out_size, void* d_ws, size_t ws_size,
                              hipStream_t stream) {
    (void)in_sizes; (void)n_in; (void)out_size; (void)ws_size;
    const float* x       = (const float*)d_in[0];
    const int*   labels  = (const int*)d_in[1];   // harness integer convention (int32)
    const float* centers = (const float*)d_in[2];
    float* out      = (float*)d_out;
    float* partials = (float*)d_ws;               // 64 floats, fully overwritten

    center_loss_partials<<<NUM_BLOCKS, WAVES_PER_BLOCK * 32, 0, stream>>>(
        x, labels, centers, partials);
    center_loss_final<<<1, 32, 0, stream>>>(partials, out);
}